// QuadraticNonLinearity_78554951844155
// MI455X (gfx1250) — compile-verified
//
#include <hip/hip_runtime.h>

typedef float v2f __attribute__((ext_vector_type(2)));
typedef float v8f __attribute__((ext_vector_type(8)));

#define N_TOT    32768
#define NCOMBO   23
#define CG_TOTAL 13455
#define FP_MAX   176    // 169 rounded up to multiple of 8 (for 2x-unrolled WMMA loop)
#define WVB      2      // waves (batch elements) per block

// ---- combo tables: (L, l1, l2) in canonical order, cumulative cg offsets ----
__constant__ int c_L[NCOMBO]   = {0,0,0,0, 2,2,2,2,2,2, 4,4,4,4,4,4,4, 6,6,6,6,6,6};
__constant__ int c_l1[NCOMBO]  = {0,2,4,6, 0,2,2,4,4,6, 0,2,2,2,4,4,6, 0,2,2,4,4,6};
__constant__ int c_l2[NCOMBO]  = {0,2,4,6, 2,2,4,4,6,6, 4,2,4,6,4,6,6, 6,4,6,4,6,6};
__constant__ int c_off[NCOMBO] = {0,1,26,107, 276,301,426,651,1056,1641,
                                  2486,2567,2792,3197,3782,4511,5564,
                                  7085,7254,7839,8684,9737,11258};
__constant__ int c_lo[4] = {0, 4, 10, 17};
__constant__ int c_hi[4] = {4, 10, 17, 23};
__constant__ long long c_outoff[4] = {0, 32768LL, 32768LL * 26, 32768LL * 107};

// =====================================================================
// CG-table setup: recompute Wigner-3j in real SH basis, on device.
// =====================================================================
struct QEnt { int row; double re, im; };

static __device__ int q_col(int l, int c, QEnt* e) {
  const double is2 = 0.70710678118654752440;
  double p = ((l >> 1) & 1) ? -1.0 : 1.0;   // (-i)^l, l even -> real
  int a = c - l;
  if (a == 0) { e[0] = {l, p, 0.0}; return 1; }
  if (a > 0) {
    double sg = (a & 1) ? -1.0 : 1.0;
    e[0] = {l - a, p * is2, 0.0};
    e[1] = {l + a, p * sg * is2, 0.0};
    return 2;
  }
  a = -a;
  double sg = (a & 1) ? -1.0 : 1.0;
  e[0] = {l - a, 0.0, -p * is2};
  e[1] = {l + a, 0.0,  p * sg * is2};
  return 2;
}

static __device__ double su2_cg(int j1, int m1, int j2, int m2, int j3, int m3,
                                const double* F) {
  if (m3 != m1 + m2) return 0.0;
  int vmin = -j1 + j2 + m3; if (-j1 + m1 > vmin) vmin = -j1 + m1; if (vmin < 0) vmin = 0;
  int vmax = j2 + j3 + m1;
  if (j3 - j1 + j2 < vmax) vmax = j3 - j1 + j2;
  if (j3 + m3 < vmax) vmax = j3 + m3;
  double C = sqrt((2.0 * j3 + 1.0) * F[j3+j1-j2] * F[j3-j1+j2] * F[j1+j2-j3]
                  * F[j3+m3] * F[j3-m3]
                  / (F[j1+j2+j3+1] * F[j1-m1] * F[j1+m1] * F[j2-m2] * F[j2+m2]));
  double S = 0.0;
  for (int v = vmin; v <= vmax; ++v) {
    double sgn = ((v + j2 + m2) & 1) ? -1.0 : 1.0;
    S += sgn * F[j2+j3+m1-v] * F[j1-m1+v]
             / (F[v] * F[j3-j1+j2-v] * F[j3+m3-v] * F[v+j1-j2-m3]);
  }
  return C * S;
}

#define SETUP_T 128
__global__ void cg_setup_kernel(float* __restrict__ ws) {
  __shared__ double Csu2[2197];
  __shared__ double Wr[2197];
  __shared__ double red[SETUP_T];

  int cb = blockIdx.x;
  int j3 = c_L[cb], j1 = c_l1[cb], j2 = c_l2[cb];
  int d1 = 2*j1 + 1, d2 = 2*j2 + 1, d3 = 2*j3 + 1;
  int M12 = d1 * d2;
  int tot = d1 * d2 * d3;

  double F[20];
  F[0] = 1.0;
  for (int t = 1; t < 20; ++t) F[t] = F[t-1] * (double)t;

  for (int t = threadIdx.x; t < tot; t += blockDim.x) {
    int i = t / (d2 * d3), k = (t / d3) % d2, nn = t % d3;
    Csu2[t] = su2_cg(j1, i - j1, j2, k - j2, j3, nn - j3, F);
  }
  __syncthreads();

  for (int t = threadIdx.x; t < tot; t += blockDim.x) {
    int jj = t / (d2 * d3), lc = (t / d3) % d2, m = t % d3;
    QEnt e1[2], e2[2], e3[2];
    int n1 = q_col(j1, jj, e1);
    int n2 = q_col(j2, lc, e2);
    int n3 = q_col(j3, m,  e3);
    double re = 0.0;
    for (int a = 0; a < n1; ++a)
      for (int b = 0; b < n2; ++b)
        for (int c = 0; c < n3; ++c) {
          double r12 = e1[a].re * e2[b].re - e1[a].im * e2[b].im;
          double i12 = e1[a].re * e2[b].im + e1[a].im * e2[b].re;
          double r   = r12 * e3[c].re + i12 * e3[c].im;  // * conj(q3), real
          re += r * Csu2[e1[a].row * (d2 * d3) + e2[b].row * d3 + e3[c].row];
        }
    Wr[t] = re;
  }
  __syncthreads();

  double ss = 0.0;
  for (int t = threadIdx.x; t < tot; t += blockDim.x) ss += Wr[t] * Wr[t];
  red[threadIdx.x] = ss;
  __syncthreads();
  for (int s = SETUP_T / 2; s > 0; s >>= 1) {
    if ((int)threadIdx.x < s) red[threadIdx.x] += red[threadIdx.x + s];
    __syncthreads();
  }
  double scale = 1.0 / sqrt(red[0]);
  if (j1 != j2) scale *= 1.41421356237309504880;

  int off = c_off[cb];
  for (int t = threadIdx.x; t < tot; t += blockDim.x) {
    int k = t / M12, rem = t % M12;
    int y = rem / d1, z = rem % d1;                  // reference reshape quirk
    float val = (float)(scale * Wr[z * (d2 * d3) + y * d3 + k]);
    ws[off + t] = val;                               // cg_r flat (k, p, q)
    int p = rem / d2, q = rem % d2;
    ws[CG_TOTAL + off + k * M12 + q * d1 + p] = val; // cg_l flat (k, q, p)
  }
}

// =====================================================================
// Main fused kernel: one wave = one batch element, blockIdx.y = L index.
// Stage 1: VALU CG contractions -> padded LDS tiles [16][FP].
// Stage 2: branchless paired ds_load_b64 fragments -> 2x chained WMMA
//          f32 16x16x4; C accumulator carries sum over combos + f-chunks.
// =====================================================================
__launch_bounds__(WVB * 32)
__global__ void quad_tp_kernel(const float* __restrict__ x0,
                               const float* __restrict__ x2,
                               const float* __restrict__ x4,
                               const float* __restrict__ x6,
                               const float* __restrict__ ws,
                               float* __restrict__ out) {
  __shared__ float xlb[WVB][16 * FP_MAX];
  __shared__ float xrb[WVB][16 * FP_MAX];
  __shared__ float in1[WVB][169];
  __shared__ float in2[WVB][169];

  const float* xs[4] = {x0, x2, x4, x6};
  int lane = threadIdx.x & 31;
  int wv   = threadIdx.x >> 5;
  int n    = blockIdx.x * WVB + wv;
  int Li   = blockIdx.y;
  int K    = 4 * Li + 1;           // 1, 5, 9, 13
  int m16  = lane & 15, h = lane >> 4;

  v8f acc = {};

  for (int cb = c_lo[Li]; cb < c_hi[Li]; ++cb) {
    int l1 = c_l1[cb], l2 = c_l2[cb];
    int d1 = 2 * l1 + 1, d2 = 2 * l2 + 1;
    int M12 = d1 * d2;
    int FP  = (M12 + 7) & ~7;      // f-dim padded to multiple of 8
    const float* cgr = ws + c_off[cb];
    const float* cgl = ws + CG_TOTAL + c_off[cb];
    const float* X1 = xs[l1 >> 1] + (long long)n * (d1 * d1);
    const float* X2 = xs[l2 >> 1] + (long long)n * (d2 * d2);

    // prefetch next combo's input tiles into cache while we compute
    if (cb + 1 < c_hi[Li]) {
      int p1 = c_l1[cb + 1], p2 = c_l2[cb + 1];
      int e1 = 2 * p1 + 1, e2 = 2 * p2 + 1;
      __builtin_prefetch(xs[p1 >> 1] + (long long)n * (e1 * e1), 0, 0);
      __builtin_prefetch(xs[p2 >> 1] + (long long)n * (e2 * e2), 0, 0);
    }

    __syncthreads();  // prev combo's stage-2 reads done before overwrite
    for (int t = lane; t < d1 * d1; t += 32) in1[wv][t] = X1[t];
    for (int t = lane; t < d2 * d2; t += 32) in2[wv][t] = X2[t];
    __syncthreads();

    // stage 1: x_l[k, p, q] and x_r[k, p, q] into padded [K][FP] layout;
    // pad columns (f >= M12) written as zero -- they feed the WMMA K-dim.
    int KF = K * FP;
    for (int t = lane; t < KF; t += 32) {
      int k = t / FP, f = t % FP;
      float sl = 0.f, sr = 0.f;
      if (f < M12) {
        int p = f / d2, q = f % d2;
        const float* cl = cgl + k * M12 + q * d1;   // cg_l[k, q, *]
        const float* cr = cgr + k * M12 + p * d2;   // cg_r[k, p, *]
        for (int j = 0; j < d1; ++j) sl += in1[wv][j * d1 + p] * cl[j];
        for (int j = 0; j < d2; ++j) sr += in2[wv][q * d2 + j] * cr[j];
      }
      xlb[wv][t] = sl;
      xrb[wv][t] = sr;
    }
    __syncthreads();

    // stage 2: 2x unrolled, branchless. Rows m>=K read stale data but only
    // pollute D rows/cols >= K, which are never stored.
    for (int c0 = 0; c0 < FP; c0 += 8) {
      int base = m16 * FP + c0 + 2 * h;             // even -> 8B aligned
      v2f a0 = *reinterpret_cast<const v2f*>(&xlb[wv][base]);
      v2f b0 = *reinterpret_cast<const v2f*>(&xrb[wv][base]);
      v2f a1 = *reinterpret_cast<const v2f*>(&xlb[wv][base + 4]);
      v2f b1 = *reinterpret_cast<const v2f*>(&xrb[wv][base + 4]);
      acc = __builtin_amdgcn_wmma_f32_16x16x4_f32(
          false, a0, false, b0, (short)0, acc, false, false);
      acc = __builtin_amdgcn_wmma_f32_16x16x4_f32(
          false, a1, false, b1, (short)0, acc, false, false);
    }
  }

  // store K x K tile: vgpr r, lane = j + 16h holds D[r+8h][j]
  long long obase = c_outoff[Li] + (long long)n * (K * K);
  if (m16 < K) {
    for (int r = 0; r < 8; ++r) {
      int mm = r + 8 * h;
      if (mm < K) out[obase + mm * K + m16] = acc[r];
    }
  }
}

// =====================================================================
extern "C" void kernel_launch(void* const* d_in, const int* in_sizes, int n_in,
                              void* d_out, int out_size, void* d_ws, size_t ws_size,
                              hipStream_t stream) {
  (void)in_sizes; (void)n_in; (void)out_size; (void)ws_size;
  const float* x0 = (const float*)d_in[0];
  const float* x2 = (const float*)d_in[1];
  const float* x4 = (const float*)d_in[2];
  const float* x6 = (const float*)d_in[3];
  float* ws = (float*)d_ws;

  cg_setup_kernel<<<NCOMBO, SETUP_T, 0, stream>>>(ws);
  quad_tp_kernel<<<dim3(N_TOT / WVB, 4), WVB * 32, 0, stream>>>(
      x0, x2, x4, x6, ws, (float*)d_out);
}